// PCFG_base_82849919140134
// MI455X (gfx1250) — compile-verified
//
#include <hip/hip_runtime.h>
#include <stdint.h>

#define NWG 16                 // B (batches), one workgroup per batch
#define NN 384                 // N
#define NTHREADS 1024          // 32 wave32 waves per workgroup
#define NWAVES 32
#define TRI ((NN * (NN - 1)) / 2)   // 73536 floats = 294144 B
#define NBUF 4                      // 4-deep diagonal prefetch ring
#define NEGV (-1000000000.0f)

// Packed strict-upper-triangle offset: (i,j) with j>i -> rowStart(i) + (j-i-1)
__device__ __forceinline__ int rowStart(int i) {
    return i * (NN - 1) - ((i * (i - 1)) >> 1);
}

// CDNA5 async global->LDS copy (GLOBAL_LOAD_ASYNC_TO_LDS_B32, tracked by ASYNCcnt).
__device__ __forceinline__ void async_ld_f32(const float* g, unsigned lds_off) {
    asm volatile("global_load_async_to_lds_b32 %0, %1, off"
                 :: "v"(lds_off), "v"(g)
                 : "memory");
}

__global__ __launch_bounds__(NTHREADS)
void cky_max_chart_kernel(const float* __restrict__ marginals,
                          float* __restrict__ out) {
    extern __shared__ float smem[];
    float* sTri = smem;            // TRI floats: per-batch chart, LDS-resident
    float* buf  = smem + TRI;      // NBUF*NN floats: prefetch ring for marg diagonal

    const int b    = blockIdx.x;
    const int tid  = threadIdx.x;
    const int lane = tid & 31;
    const int wave = tid >> 5;

    const float* mb = marginals + (size_t)b * NN * NN;
    float* s_out = out + (size_t)b * NN * NN;
    float* p_out = out + (size_t)NWG * NN * NN + (size_t)b * NN * NN;

    // Prologue: async-prefetch marginal diagonals for widths 2 and 3.
    if (tid < NN - 2)
        async_ld_f32(mb + (size_t)tid * (NN + 1) + 2,
                     (unsigned)(size_t)&buf[(2 & (NBUF - 1)) * NN + tid]);
    if (tid < NN - 3)
        async_ld_f32(mb + (size_t)tid * (NN + 1) + 3,
                     (unsigned)(size_t)&buf[(3 & (NBUF - 1)) * NN + tid]);

    // Init: s = NEG on/below diagonal, width-1 diag = marginals; p = 0 for j<=i+1.
    // DP below writes exactly the disjoint j>=i+2 region -> no global ordering hazard.
    for (int idx = tid; idx < NN * NN; idx += NTHREADS) {
        int i = idx / NN;
        int j = idx - i * NN;
        if (j <= i)     s_out[idx] = NEGV;
        if (j <= i + 1) p_out[idx] = 0.0f;
        if (j == i + 1) {
            float m = mb[idx];
            s_out[idx] = m;
            sTri[rowStart(i)] = m;   // width-1 chart entries
        }
    }

    for (int w = 2; w < NN; ++w) {
        // Per-wave: oldest in-flight async load (width w's diagonal) must be done.
        if (w + 1 < NN) asm volatile("s_wait_asynccnt 1" ::: "memory");
        else            asm volatile("s_wait_asynccnt 0" ::: "memory");
        // Single barrier per width: publishes buf[w] completions + width w-1 chart.
        __syncthreads();

        const float* mcur = buf + (w & (NBUF - 1)) * NN;
        const int cells = NN - w;

        if (cells >= 128) {
            // Thread-per-cell: serial k-loop, no reduction traffic.
            int i = tid;
            if (i < cells) {
                const int rsi = rowStart(i);
                float best  = -3.402823e38f;
                int   bestk = 0;
                int rz = rowStart(i + 1) + (w - 2);   // addr of s[i+1, i+w]
                for (int k = 0; k < w - 1; ++k) {
                    float y = sTri[rsi + k];          // s[i, i+1+k]
                    float z = sTri[rz];               // s[i+1+k, i+w]
                    float v = y + z;
                    if (v > best) { best = v; bestk = k; }   // strict > = first argmax
                    rz += (NN - 3 - i - k);           // incremental triangle address
                }
                float val = best + mcur[i];
                sTri[rsi + (w - 1)] = val;
                int j = i + w;
                s_out[i * NN + j] = val;
                p_out[i * NN + j] = (float)(bestk + i + 1);
            }
        } else {
            // Wave-per-cell: lanes parallelize the (large) split dimension k.
            for (int i = wave; i < cells; i += NWAVES) {
                const int rsi = rowStart(i);
                float best  = -3.402823e38f;
                int   bestk = 0x7fffffff;
                for (int k0 = 0; k0 < w - 1; k0 += 32) {
                    int k = k0 + lane;
                    if (k < w - 1) {
                        float y = sTri[rsi + k];
                        float z = sTri[rowStart(i + 1 + k) + (w - 2 - k)];
                        float v = y + z;
                        if (v > best) { best = v; bestk = k; }
                    }
                }
                #pragma unroll
                for (int off = 16; off > 0; off >>= 1) {
                    float ov = __shfl_xor(best, off, 32);
                    int   ok = __shfl_xor(bestk, off, 32);
                    if (ov > best || (ov == best && ok < bestk)) { best = ov; bestk = ok; }
                }
                if (lane == 0) {
                    float val = best + mcur[i];
                    sTri[rsi + (w - 1)] = val;
                    int j = i + w;
                    s_out[i * NN + j] = val;
                    p_out[i * NN + j] = (float)(bestk + i + 1);
                }
            }
        }

        // Prefetch width w+2 into ring slot (w+2)&3. Safe without a second barrier:
        // concurrent waves are all past barrier w, so slot (w+2)&3 (last read at
        // width w-2) has no remaining readers; slots w&3 (reading) and (w+1)&3
        // (in flight) are distinct mod 4.
        const int wn = w + 2;
        if (wn < NN && tid < NN - wn)
            async_ld_f32(mb + (size_t)tid * (NN + 1) + wn,
                         (unsigned)(size_t)&buf[(wn & (NBUF - 1)) * NN + tid]);
    }
}

extern "C" void kernel_launch(void* const* d_in, const int* in_sizes, int n_in,
                              void* d_out, int out_size, void* d_ws, size_t ws_size,
                              hipStream_t stream) {
    (void)in_sizes; (void)n_in; (void)out_size; (void)d_ws; (void)ws_size;
    const float* marginals = (const float*)d_in[0];
    // d_in[1] (lens) is unused by the reference computation.
    float* out = (float*)d_out;
    const size_t smem_bytes = (size_t)(TRI + NBUF * NN) * sizeof(float); // ~300 KB < 320 KB
    cky_max_chart_kernel<<<dim3(NWG), dim3(NTHREADS), smem_bytes, stream>>>(marginals, out);
}